// SelectiveScanModule_6828998001032
// MI455X (gfx1250) — compile-verified
//
#include <hip/hip_runtime.h>

// Mamba selective scan, chunked semiseparable formulation on gfx1250.
// One wave32 per (b,d); 128 chunks of 16 timesteps; intra-chunk 16x16x16
// f32 matmul via 4 chained v_wmma_f32_16x16x4_f32; state carried in VGPRs.
// Cross-lane reductions use DPP (row_shr / row_ror) + v_permlanex16 instead
// of ds_bpermute to keep the reduction network on the VALU dual-issue path.

#define BATCH   4
#define DIN     2048
#define NSTATE  16
#define SEQLEN  2048
#define CHUNK   16
#define NCHUNK  (SEQLEN / CHUNK)

typedef float v2f __attribute__((ext_vector_type(2)));
typedef float v8f __attribute__((ext_vector_type(8)));

// DPP move: returns src permuted by CTRL; invalid source lanes -> 0.
template <int CTRL>
__device__ __forceinline__ float dpp_mov(float x) {
  int r = __builtin_amdgcn_update_dpp(0, __builtin_bit_cast(int, x),
                                      CTRL, 0xF, 0xF, true);
  return __builtin_bit_cast(float, r);
}

// Sum of all 16 lanes within each DPP row (lanes 0-15 / 16-31), replicated.
__device__ __forceinline__ float allreduce16(float v) {
  v += dpp_mov<0x121>(v);  // row_ror:1
  v += dpp_mov<0x122>(v);  // row_ror:2
  v += dpp_mov<0x124>(v);  // row_ror:4
  v += dpp_mov<0x128>(v);  // row_ror:8
  return v;
}

// Inclusive prefix sum within each DPP row (Hillis-Steele, zero fill).
__device__ __forceinline__ float prefix16(float v) {
  v += dpp_mov<0x111>(v);  // row_shr:1
  v += dpp_mov<0x112>(v);  // row_shr:2
  v += dpp_mov<0x114>(v);  // row_shr:4
  v += dpp_mov<0x118>(v);  // row_shr:8
  return v;
}

// Swap the two 16-lane halves (lane l <-> lane l^16) via v_permlanex16_b32.
__device__ __forceinline__ float xhalf16(float v) {
  int i = __builtin_bit_cast(int, v);
  int r = __builtin_amdgcn_permlanex16(i, i, 0x76543210, 0xFEDCBA98,
                                       false, false);
  return __builtin_bit_cast(float, r);
}

__device__ __forceinline__ float softplus_f(float x) {
  return (x > 20.0f) ? x : __logf(1.0f + __expf(x));
}

__global__ __launch_bounds__(128) void mamba_scan_kernel(
    const float* __restrict__ u,     // (b, d, l)
    const float* __restrict__ delta, // (b, d, l)
    const float* __restrict__ A,     // (d, n)
    const float* __restrict__ Bm,    // (b, n, l)
    const float* __restrict__ Cm,    // (b, n, l)
    const float* __restrict__ Dv,    // (d,)
    const float* __restrict__ z,     // (b, d, l)
    float* __restrict__ out)         // (b, d, l)
{
  const int lane  = threadIdx.x & 31;
  const int wave  = (blockIdx.x << 2) | (threadIdx.x >> 5); // global wave id
  const int b     = wave / DIN;
  const int d     = wave % DIN;

  const int tl    = lane & 15;   // timestep-in-chunk / column index
  const int half  = lane >> 4;   // 0: lanes 0-15, 1: lanes 16-31

  // Per-lane A decay factors for the 8 state rows this lane touches in the
  // WMMA operand layout: n(q,j) = 4q + j + 2*half  (operand VGPR pair holds
  // K-rows {0,2} low/high half in VGPR0 and {1,3} in VGPR1).
  float Afac[4][2];
#pragma unroll
  for (int q = 0; q < 4; ++q)
#pragma unroll
    for (int j = 0; j < 2; ++j)
      Afac[q][j] = A[(size_t)d * NSTATE + (4 * q + j + 2 * half)];

  const float Dd = Dv[d];

  const size_t rowU  = ((size_t)b * DIN + d) * SEQLEN;   // u/delta/z/out row
  const size_t rowBC = (size_t)b * NSTATE * SEQLEN;      // B/C batch base

  // Recurrent state x[n], distributed over the operand layout (8 rows per
  // lane; replicated across each 16-lane half after the allreduce).
  float x[4][2];
#pragma unroll
  for (int q = 0; q < 4; ++q) { x[q][0] = 0.0f; x[q][1] = 0.0f; }

  for (int c = 0; c < NCHUNK; ++c) {
    const int base = c * CHUNK;

    // Prefetch next chunk's streaming rows (lowers to global_prefetch_b8).
    if (c + 1 < NCHUNK) {
      __builtin_prefetch(&u[rowU + base + CHUNK + tl], 0, 3);
      __builtin_prefetch(&delta[rowU + base + CHUNK + tl], 0, 3);
      __builtin_prefetch(&z[rowU + base + CHUNK + tl], 0, 3);
    }

    const float del = delta[rowU + base + tl];
    const float uu  = u[rowU + base + tl];
    const float zz  = z[rowU + base + tl];

    const float dtv = softplus_f(del);
    const float du  = dtv * uu;

    // Inclusive prefix sum of dt within each half -> s_t; S = chunk total.
    const float sT = prefix16(dtv);
    const float S  = __shfl(sT, 15, 16);

    // Build WMMA operands:
    //   a[q]  = CT slice (A-matrix 16x4): lane tl = row t, K-cols n=4q+j+2h
    //   bm[q] = BU' slice (B-matrix 4x16): lane tl = col s, K-rows n=4q+j+2h
    // Carried state x_in is folded into column s==0 of BU'.
    v2f a[4], bm[4];
#pragma unroll
    for (int q = 0; q < 4; ++q) {
#pragma unroll
      for (int j = 0; j < 2; ++j) {
        const int    n   = 4 * q + j + 2 * half;
        const size_t idx = rowBC + (size_t)n * SEQLEN + base + tl;
        const float  cr  = Cm[idx];
        const float  br  = Bm[idx];
        const float  af  = Afac[q][j];
        float ct = cr * __expf(af * sT);
        float bu = br * du * __expf(-af * sT);
        if (tl == 0) bu += x[q][j];
        a[q][j]  = ct;
        bm[q][j] = bu;
      }
    }

    // K(16x16) = CT(16x16) x BU'^T via 4 chained f32 WMMAs (K-dim 4 each).
    v8f K = {0.f, 0.f, 0.f, 0.f, 0.f, 0.f, 0.f, 0.f};
#pragma unroll
    for (int q = 0; q < 4; ++q) {
      K = __builtin_amdgcn_wmma_f32_16x16x4_f32(
          false, a[q], false, bm[q], (short)0, K, false, false);
    }

    // State update: x_out[n] = exp(A[n]*S) * sum_s BU'[s,n]
    // (allreduce over each 16-lane half = sum over columns s).
#pragma unroll
    for (int q = 0; q < 4; ++q) {
#pragma unroll
      for (int j = 0; j < 2; ++j) {
        const float s = allreduce16(bm[q][j]);
        x[q][j] = s * __expf(Afac[q][j] * S);
      }
    }

    // y_t = sum_{s<=t} K[t,s].  D layout: VGPR r holds rows {r, r+8} in the
    // two lane halves; lane column = tl.  Mask upper triangle, then rowsum.
    float y[8];
#pragma unroll
    for (int r = 0; r < 8; ++r) {
      const int row = r + 8 * half;
      const float v = (tl <= row) ? K[r] : 0.0f;
      y[r] = allreduce16(v);
    }

    // Gather: lane tl (low half) ends up holding y_{tl}.
    float Y = y[0];
#pragma unroll
    for (int r = 0; r < 8; ++r) {
      const float hiv = xhalf16(y[r]);   // bring high-half rows down
      Y = (tl == r) ? y[r] : Y;
      Y = (tl == r + 8) ? hiv : Y;
    }

    // Epilogue: skip connection + SiLU(z) gate, store 16 contiguous floats.
    float yo  = Y + Dd * uu;
    const float sig = 1.0f / (1.0f + __expf(-zz));
    yo *= zz * sig;
    if (lane < 16) out[rowU + base + tl] = yo;
  }
}

extern "C" void kernel_launch(void* const* d_in, const int* in_sizes, int n_in,
                              void* d_out, int out_size, void* d_ws, size_t ws_size,
                              hipStream_t stream) {
  (void)in_sizes; (void)n_in; (void)d_ws; (void)ws_size; (void)out_size;
  const float* u     = (const float*)d_in[0];
  const float* delta = (const float*)d_in[1];
  const float* A     = (const float*)d_in[2];
  const float* Bm    = (const float*)d_in[3];
  const float* Cm    = (const float*)d_in[4];
  const float* Dv    = (const float*)d_in[5];
  const float* z     = (const float*)d_in[6];
  float* out         = (float*)d_out;

  // One wave32 per (b,d): 8192 waves, 4 waves per 128-thread block.
  const int nWaves = BATCH * DIN;
  dim3 grid(nWaves / 4), block(128);
  mamba_scan_kernel<<<grid, block, 0, stream>>>(u, delta, A, Bm, Cm, Dv, z, out);
}